// E3Selfmix_29609504538680
// MI455X (gfx1250) — compile-verified
//
#include <hip/hip_runtime.h>
#include <stdint.h>

typedef __attribute__((ext_vector_type(2))) float v2f;
typedef __attribute__((ext_vector_type(4))) float v4f;
typedef __attribute__((ext_vector_type(8))) float v8f;

#if __has_builtin(__builtin_amdgcn_wmma_f32_16x16x4_f32)
#define HAVE_WMMA_F32X4 1
#else
#define HAVE_WMMA_F32X4 0
#endif

// ---------------------------------------------------------------------------
// Compile-time table of the 45 symmetric monomial pairs (i<=j), padded to 48.
// feat[f] = x[pi[f]] * x[pj[f]];  Wfused folds the (i,j)+(j,i) symmetry.
// ---------------------------------------------------------------------------
struct PairTab { int pi[48]; int pj[48]; };
static constexpr PairTab mkPairs() {
    PairTab t{};
    int f = 0;
    for (int i = 0; i < 9; ++i)
        for (int j = i; j < 9; ++j) { t.pi[f] = i; t.pj[f] = j; ++f; }
    for (; f < 48; ++f) { t.pi[f] = 0; t.pj[f] = 0; }
    return t;
}
static constexpr PairTab PT = mkPairs();

// ---------------------------------------------------------------------------
// Prep kernel: build Wfused[16][48] = sum_paths w_path * norm_l3 * realCG
// Runs once, 1 block. Deterministic (fixed-order reduction, no atomics).
// ---------------------------------------------------------------------------
__device__ static double dfact(int n) {
    double r = 1.0;
    for (int i = 2; i <= n; ++i) r *= (double)i;
    return r;
}

// Complex-basis Clebsch-Gordan <j1 m1 j2 m2 | j3 m3> (Racah formula).
__device__ static double cgc(int j1, int m1, int j2, int m2, int j3, int m3) {
    if (m1 + m2 != m3) return 0.0;
    double pref = sqrt((2.0 * j3 + 1.0) * dfact(j3 + j1 - j2) * dfact(j3 - j1 + j2) *
                       dfact(j1 + j2 - j3) / dfact(j1 + j2 + j3 + 1));
    pref *= sqrt(dfact(j3 + m3) * dfact(j3 - m3) * dfact(j1 - m1) * dfact(j1 + m1) *
                 dfact(j2 - m2) * dfact(j2 + m2));
    double s = 0.0;
    for (int k = 0; k <= j1 + j2 - j3; ++k) {
        int d1 = j1 + j2 - j3 - k, d2 = j1 - m1 - k, d3 = j2 + m2 - k;
        int d4 = j3 - j2 + m1 + k, d5 = j3 - j1 - m2 + k;
        if (d1 < 0 || d2 < 0 || d3 < 0 || d4 < 0 || d5 < 0) continue;
        double den = dfact(k) * dfact(d1) * dfact(d2) * dfact(d3) * dfact(d4) * dfact(d5);
        s += ((k & 1) ? -1.0 : 1.0) / den;
    }
    return pref * s;
}

// Complex -> real spherical-harmonic change of basis (m = -l..l ordering).
__device__ static void c2r(int l, double* Qr, double* Qi) {
    int n = 2 * l + 1;
    for (int i = 0; i < n * n; ++i) { Qr[i] = 0.0; Qi[i] = 0.0; }
    Qr[l * n + l] = 1.0;
    double is2 = 1.0 / sqrt(2.0);
    for (int m = 1; m <= l; ++m) {
        double sgn = (m & 1) ? -1.0 : 1.0;
        Qr[(l + m) * n + (l + m)] = sgn * is2;
        Qr[(l + m) * n + (l - m)] = is2;
        Qi[(l - m) * n + (l - m)] = is2;
        Qi[(l - m) * n + (l + m)] = -sgn * is2;
    }
}

__global__ void e3_prep(const float* __restrict__ w0, const float* __restrict__ w1,
                        const float* __restrict__ w2, const float* __restrict__ w3,
                        float* __restrict__ Wg) {
    __shared__ float Wp[13][768];
    const int t = threadIdx.x;
    for (int p = 0; p < 13; ++p)
        for (int i = t; i < 768; i += blockDim.x) Wp[p][i] = 0.f;
    __syncthreads();

    if (t < 13) {
        // Paths in the reference's order; weight index = position within l3's list.
        const int pl1[13] = {0, 1, 2,  0, 1, 1, 2,  0, 1, 2, 2,  1, 2};
        const int pl2[13] = {0, 1, 2,  1, 0, 2, 1,  2, 1, 0, 2,  2, 1};
        const int pl3[13] = {0, 0, 0,  1, 1, 1, 1,  2, 2, 2, 2,  3, 3};
        const int pwi[13] = {0, 1, 2,  0, 1, 2, 3,  0, 1, 2, 3,  0, 1};
        const int xoff[3] = {0, 1, 4};
        const int yoff[4] = {0, 1, 4, 9};

        const int l1 = pl1[t], l2 = pl2[t], l3 = pl3[t];
        const int n1 = 2 * l1 + 1, n2 = 2 * l2 + 1, n3 = 2 * l3 + 1;
        double w, nrm;
        switch (l3) {
            case 0:  w = (double)w0[pwi[t]]; nrm = 1.0 / sqrt(3.0); break;
            case 1:  w = (double)w1[pwi[t]]; nrm = 0.5;             break;
            case 2:  w = (double)w2[pwi[t]]; nrm = 0.5;             break;
            default: w = (double)w3[pwi[t]]; nrm = 1.0 / sqrt(2.0); break;
        }

        double Q1r[25], Q1i[25], Q2r[25], Q2i[25], Q3r[49], Q3i[49];
        c2r(l1, Q1r, Q1i);
        c2r(l2, Q2r, Q2i);
        c2r(l3, Q3r, Q3i);

        double CG[175];  // max 5*5*7
        for (int a = 0; a < n1; ++a)
            for (int b = 0; b < n2; ++b)
                for (int c = 0; c < n3; ++c)
                    CG[(a * n2 + b) * n3 + c] = cgc(l1, a - l1, l2, b - l2, l3, c - l3);

        // X[a][b][c] = Re( Q1[a][m] Q2[b][n] conj(Q3[c][o]) CG[m][n][o] )
        for (int a = 0; a < n1; ++a)
            for (int b = 0; b < n2; ++b)
                for (int c = 0; c < n3; ++c) {
                    double xr = 0.0;
                    for (int m = 0; m < n1; ++m)
                        for (int n = 0; n < n2; ++n)
                            for (int o = 0; o < n3; ++o) {
                                double cg = CG[(m * n2 + n) * n3 + o];
                                if (cg == 0.0) continue;
                                double r1 = Q1r[a * n1 + m], i1 = Q1i[a * n1 + m];
                                double r2 = Q2r[b * n2 + n], i2 = Q2i[b * n2 + n];
                                double tr = r1 * r2 - i1 * i2, ti = r1 * i2 + i1 * r2;
                                double r3 = Q3r[c * n3 + o], i3 = -Q3i[c * n3 + o];
                                xr += (tr * r3 - ti * i3) * cg;
                            }
                    double val = xr * w * nrm;
                    if (val != 0.0) {
                        int i = xoff[l1] + a, j = xoff[l2] + b, mrow = yoff[l3] + c;
                        int ii = i < j ? i : j, jj = i < j ? j : i;
                        int f = 9 * ii - (ii * (ii - 1)) / 2 + (jj - ii);
                        Wp[t][mrow * 48 + f] += (float)val;
                    }
                }
    }
    __syncthreads();
    for (int i = t; i < 768; i += blockDim.x) {
        float s = 0.f;
        for (int p = 0; p < 13; ++p) s += Wp[p][i];
        Wg[i] = s;
    }
}

// ---------------------------------------------------------------------------
// Main kernel: 256 threads = 8 waves; each wave does one 16-element WMMA tile
// (128 elements / block). x tile staged into LDS via async global->LDS b128.
//   y(16 x 16elem) = sum over 12 K-chunks  A(16x4 Wfused) * B(4x16 feats)
// using V_WMMA_F32_16X16X4_F32; D layout makes stores 2x b128, fully coalesced.
// ---------------------------------------------------------------------------
__global__ __launch_bounds__(256) void e3_main(const float* __restrict__ x,
                                               const float* __restrict__ Wg,
                                               const float* __restrict__ b0,
                                               const float* __restrict__ kres,
                                               float* __restrict__ out,
                                               long long total, int C) {
    __shared__ float sx[128 * 9];    // 4608 B staged x tile
    __shared__ float sW[16 * 52];    // Wfused, stride 52 -> conflict-free reads

    const int t = threadIdx.x;
    const long long blockBase = (long long)blockIdx.x * 128;

    // Stage Wfused (3 KB) into LDS.
    for (int i = t; i < 768; i += 256) sW[(i / 48) * 52 + (i % 48)] = Wg[i];

    // Async-stage the x tile: CDNA5 global->LDS async copy (ASYNCcnt tracked).
    {
        long long elemsLeft = total - blockBase;
        int elems = elemsLeft >= 128 ? 128 : (elemsLeft > 0 ? (int)elemsLeft : 0);
        int bytes = elems * 36;
        int fullChunks = bytes >> 4;  // whole 16B chunks (36*128 = 4608 = 288*16)
        const float* gsrc = x + blockBase * 9;
        unsigned long long sbase = (unsigned long long)(uintptr_t)gsrc;
        unsigned ldsBase = (unsigned)(uintptr_t)&sx[0];
        for (int c = t; c < fullChunks; c += 256) {
            unsigned ldsAddr = ldsBase + (unsigned)(c << 4);
            int byteOff = c << 4;
            asm volatile("global_load_async_to_lds_b128 %0, %1, %2"
                         :: "v"(ldsAddr), "v"(byteOff), "s"(sbase) : "memory");
        }
        // Ragged tail (only when elems < 128): plain loads.
        for (int i = (fullChunks << 2) + t; i < (bytes >> 2); i += 256) sx[i] = gsrc[i];
        asm volatile("s_wait_asynccnt 0" ::: "memory");
    }
    __syncthreads();

    const int lane = t & 31;
    const int le = lane & 15;       // element within the wave's 16-element tile
    const bool hi = lane >= 16;     // high half-wave: K rows {2,3}, M rows 8..15
    const int eLocal = (t >> 5) * 16 + le;
    const long long e = blockBase + eLocal;

    float xv[9];
#pragma unroll
    for (int i = 0; i < 9; ++i) xv[i] = sx[eLocal * 9 + i];

    // 45 symmetric quadratic monomials (compile-time indices after unroll).
    float feat[45];
#pragma unroll
    for (int f = 0; f < 45; ++f) feat[f] = xv[PT.pi[f]] * xv[PT.pj[f]];

    const int row = le;                 // A-matrix row M carried by this lane
    const int colSel = hi ? 2 : 0;      // K split between half-waves

    float y[8];
#if HAVE_WMMA_F32X4
    v8f acc = {0.f, 0.f, 0.f, 0.f, 0.f, 0.f, 0.f, 0.f};
#pragma unroll
    for (int c = 0; c < 12; ++c) {
        v2f A;
        A.x = sW[row * 52 + c * 4 + colSel + 0];
        A.y = sW[row * 52 + c * 4 + colSel + 1];
        float bl0 = feat[c * 4 + 0];
        float bl1 = (c * 4 + 1 < 45) ? feat[c * 4 + 1] : 0.f;
        float bh0 = (c * 4 + 2 < 45) ? feat[c * 4 + 2] : 0.f;
        float bh1 = (c * 4 + 3 < 45) ? feat[c * 4 + 3] : 0.f;
        v2f B;
        B.x = hi ? bh0 : bl0;
        B.y = hi ? bh1 : bl1;
        // D = A(16x4) * B(4x16) + C   (f32 WMMA, accumulate across K-chunks)
        acc = __builtin_amdgcn_wmma_f32_16x16x4_f32(false, A, false, B,
                                                    (short)0, acc, false, false);
    }
#pragma unroll
    for (int i = 0; i < 8; ++i) y[i] = acc[i];
#else
    // Fallback: dense dot products (keeps file compiling if builtin absent).
#pragma unroll
    for (int i = 0; i < 8; ++i) {
        int m = (hi ? 8 : 0) + i;
        float s = 0.f;
        for (int f = 0; f < 45; ++f) s += sW[m * 52 + f] * feat[f];
        y[i] = s;
    }
#endif

    // Bias on scalars + gated residual on 0e/1o/2e (3o passes through).
    const int ch = (int)(e % (long long)C);
    const float kk0 = kres[ch * 3 + 0];
    const float kk1 = kres[ch * 3 + 1];
    const float kk2 = kres[ch * 3 + 2];
    if (!hi) {  // outputs m = 0..7
        y[0] += b0[0] + kk0 * xv[0];
        y[1] += kk1 * xv[1]; y[2] += kk1 * xv[2]; y[3] += kk1 * xv[3];
        y[4] += kk2 * xv[4]; y[5] += kk2 * xv[5]; y[6] += kk2 * xv[6]; y[7] += kk2 * xv[7];
    } else {    // outputs m = 8..15 ; only m=8 (last 2e component) has residual
        y[0] += kk2 * xv[8];
    }

    if (e < total) {
        float* op = out + e * 16 + (hi ? 8 : 0);
        v4f v0 = {y[0], y[1], y[2], y[3]};
        v4f v1 = {y[4], y[5], y[6], y[7]};
        *(v4f*)op = v0;
        *(v4f*)(op + 4) = v1;
    }
}

// ---------------------------------------------------------------------------
extern "C" void kernel_launch(void* const* d_in, const int* in_sizes, int n_in,
                              void* d_out, int out_size, void* d_ws, size_t ws_size,
                              hipStream_t stream) {
    const float* x  = (const float*)d_in[0];
    const float* w0 = (const float*)d_in[1];
    const float* w1 = (const float*)d_in[2];
    const float* w2 = (const float*)d_in[3];
    const float* w3 = (const float*)d_in[4];
    const float* b0 = (const float*)d_in[5];
    const float* kk = (const float*)d_in[6];
    float* Wg = (float*)d_ws;  // 16*48 floats of scratch for Wfused

    long long total = (long long)in_sizes[0] / 9;  // B*C elements
    int C = in_sizes[6] / 3;                       // channels (k is (C,3))
    if (C <= 0) C = 1;

    e3_prep<<<1, 64, 0, stream>>>(w0, w1, w2, w3, Wg);

    int blocks = (int)((total + 127) / 128);
    e3_main<<<blocks, 256, 0, stream>>>(x, Wg, b0, kk, (float*)d_out, total, C);
}